// CrystalDiffusionModel_28355374088954
// MI455X (gfx1250) — compile-verified
//
#include <hip/hip_runtime.h>
#include <math.h>

#define BB 2
#define NN 384
#define NDIM 8
#define CDIM 16
#define HH 128
#define LL 4
#define TSTEPS 100
#define BN (BB * NN)

typedef float v2f __attribute__((ext_vector_type(2)));
typedef float v8f __attribute__((ext_vector_type(8)));

__device__ __forceinline__ float silu_f(float x) { return x / (1.0f + __expf(-x)); }

// ---------------- f32 WMMA GEMM epilogue modes ----------------
#define EPI_NONE 0        // C = acc
#define EPI_BIAS 1        // C = acc + bias[n]
#define EPI_BIAS_SILU 2   // C = silu(acc + bias[n])
#define EPI_ACC_SILU 3    // C = silu(acc + Cin)
#define EPI_EDGE_AGG 4    // C = (acc + bias[n]*rs[m]) / max(rs[m],1)
#define EPI_ADD_SCALED 5  // C = Cin + (acc + bias[n]) * rs[m]
#define EPI_BIAS_ADDB 6   // C = acc + bias[n] + rs[(m/NN)*Nc + n]   (per-batch vector add)

// ---------------- full-tile WMMA GEMM -------------------------------------------------
// 4 waves per block; wave w computes the 16x32 tile at columns [w*32, w*32+32).
// Requires: M % 16 == 0, K % 4 == 0, Nc == 128. No per-lane predication anywhere.
template <int EPI>
__global__ void gemm16x32_wmma(const float* __restrict__ A, const float* __restrict__ W,
                               float* __restrict__ C, const float* __restrict__ bias,
                               const float* __restrict__ Cin, const float* __restrict__ rs,
                               int M, int K, int Nc) {
  const int lane = threadIdx.x & 31;
  const int wid  = threadIdx.x >> 5;  // 0..3
  const int half = lane >> 4;         // 0 or 1
  const int nm   = lane & 15;         // 0..15
  const int tm = blockIdx.x * 16;
  const int tn = wid * 32;
  const int arow = tm + nm;       // A row this lane loads (M = nm)
  const int col0 = tn + nm;       // first  16-col tile
  const int col1 = tn + 16 + nm;  // second 16-col tile

  v8f acc0 = {0.f, 0.f, 0.f, 0.f, 0.f, 0.f, 0.f, 0.f};
  v8f acc1 = {0.f, 0.f, 0.f, 0.f, 0.f, 0.f, 0.f, 0.f};
  const float* __restrict__ ap = A + (size_t)arow * K;
#pragma unroll 4
  for (int k = 0; k < K; k += 4) {
    const int ka = k + 2 * half;  // lanes 0-15: K0,K1 ; lanes 16-31: K2,K3
    v2f a, b0, b1;
    a.x = ap[ka];
    a.y = ap[ka + 1];
    const float* __restrict__ wp = W + (size_t)ka * Nc;
    b0.x = wp[col0];
    b1.x = wp[col1];
    b0.y = wp[Nc + col0];
    b1.y = wp[Nc + col1];
    acc0 = __builtin_amdgcn_wmma_f32_16x16x4_f32(false, a, false, b0, (short)0, acc0,
                                                 false, false);
    acc1 = __builtin_amdgcn_wmma_f32_16x16x4_f32(false, a, false, b1, (short)0, acc1,
                                                 false, false);
  }

  float bcol0 = 0.f, bcol1 = 0.f;
  if (EPI == EPI_BIAS || EPI == EPI_BIAS_SILU || EPI == EPI_EDGE_AGG ||
      EPI == EPI_ADD_SCALED || EPI == EPI_BIAS_ADDB) {
    bcol0 = bias[col0];
    bcol1 = bias[col1];
  }

#pragma unroll
  for (int r = 0; r < 8; ++r) {
    const int orow = tm + r + 8 * half;  // C/D layout: M = r + 8*half, N = nm
    const size_t i0 = (size_t)orow * Nc + col0;
    const size_t i1 = (size_t)orow * Nc + col1;
    const float v0 = acc0[r];
    const float v1 = acc1[r];
    if (EPI == EPI_NONE) {
      C[i0] = v0;
      C[i1] = v1;
    } else if (EPI == EPI_BIAS) {
      C[i0] = v0 + bcol0;
      C[i1] = v1 + bcol1;
    } else if (EPI == EPI_BIAS_SILU) {
      C[i0] = silu_f(v0 + bcol0);
      C[i1] = silu_f(v1 + bcol1);
    } else if (EPI == EPI_ACC_SILU) {
      C[i0] = silu_f(v0 + Cin[i0]);
      C[i1] = silu_f(v1 + Cin[i1]);
    } else if (EPI == EPI_EDGE_AGG) {
      const float c = rs[orow];
      const float inv = 1.0f / fmaxf(c, 1.0f);
      C[i0] = (v0 + bcol0 * c) * inv;
      C[i1] = (v1 + bcol1 * c) * inv;
    } else if (EPI == EPI_ADD_SCALED) {
      const float s = rs[orow];
      C[i0] = Cin[i0] + (v0 + bcol0) * s;
      C[i1] = Cin[i1] + (v1 + bcol1) * s;
    } else if (EPI == EPI_BIAS_ADDB) {
      const int bb = orow / NN;
      C[i0] = v0 + bcol0 + rs[bb * Nc + col0];
      C[i1] = v1 + bcol1 + rs[bb * Nc + col1];
    }
  }
}

// ---------------- guarded single-tile WMMA GEMM for narrow outputs (Nc < 16) ----------
template <int EPI>
__global__ void gemm16_narrow_wmma(const float* __restrict__ A, const float* __restrict__ W,
                                   float* __restrict__ C, const float* __restrict__ bias,
                                   int M, int K, int Nc) {
  const int lane = threadIdx.x & 31;
  const int half = lane >> 4;
  const int nm   = lane & 15;
  const int tm = blockIdx.x * 16;
  const int arow = tm + nm;
  const int col  = nm;
  const bool colok = (col < Nc);

  v8f acc = {0.f, 0.f, 0.f, 0.f, 0.f, 0.f, 0.f, 0.f};
  const float* __restrict__ ap = A + (size_t)arow * K;
  for (int k = 0; k < K; k += 4) {
    const int ka = k + 2 * half;
    v2f a, b;
    a.x = ap[ka];
    a.y = ap[ka + 1];
    b.x = colok ? W[(size_t)ka * Nc + col] : 0.f;
    b.y = colok ? W[(size_t)(ka + 1) * Nc + col] : 0.f;
    acc = __builtin_amdgcn_wmma_f32_16x16x4_f32(false, a, false, b, (short)0, acc,
                                                false, false);
  }
  const float bcol = colok ? bias[col] : 0.f;
#pragma unroll
  for (int r = 0; r < 8; ++r) {
    const int orow = tm + r + 8 * half;
    if (!colok) continue;
    C[(size_t)orow * Nc + col] = acc[r] + bcol;
  }
}

// ---------------- small per-batch prep: alpha_bar, time+cond embedding MLPs ----------------
__global__ void prep_small_kernel(const int* __restrict__ timesteps,
                                  const float* __restrict__ condition,
                                  const float* __restrict__ time_w1, const float* __restrict__ time_b1,
                                  const float* __restrict__ time_w2, const float* __restrict__ time_b2,
                                  const float* __restrict__ cond_w1, const float* __restrict__ cond_b1,
                                  const float* __restrict__ cond_w2, const float* __restrict__ cond_b2,
                                  float* __restrict__ sab, float* __restrict__ tcvec) {
  const int b = blockIdx.x;
  const int h = threadIdx.x;  // 0..127
  __shared__ float sx[HH];
  __shared__ float sh[HH];

  const int t = timesteps[b];
  // betas = linspace(1e-4, 0.02, 100); alpha_bar = cumprod(1-beta)[t]
  float abar = 1.0f;
  const float step = (0.02f - 1e-4f) / (float)(TSTEPS - 1);
  for (int k = 0; k <= t; ++k) abar *= (1.0f - (1e-4f + step * (float)k));
  if (h == 0) {
    sab[2 * b + 0] = sqrtf(abar);
    sab[2 * b + 1] = sqrtf(1.0f - abar);
  }

  // sinusoidal time embedding
  const float factor = logf(10000.0f) / (float)(HH / 2 - 1);
  const float tf = (float)t;
  sx[h] = (h < HH / 2) ? sinf(tf * __expf(-factor * (float)h))
                       : cosf(tf * __expf(-factor * (float)(h - HH / 2)));
  __syncthreads();

  float a1 = time_b1[h];
  for (int k = 0; k < HH; ++k) a1 += sx[k] * time_w1[k * HH + h];
  __syncthreads();
  sh[h] = silu_f(a1);
  __syncthreads();
  float tv = time_b2[h];
  for (int k = 0; k < HH; ++k) tv += sh[k] * time_w2[k * HH + h];
  __syncthreads();

  if (h < CDIM) sx[h] = condition[b * CDIM + h];
  __syncthreads();
  float a2 = cond_b1[h];
  for (int k = 0; k < CDIM; ++k) a2 += sx[k] * cond_w1[k * HH + h];
  __syncthreads();
  sh[h] = silu_f(a2);
  __syncthreads();
  float cv = cond_b2[h];
  for (int k = 0; k < HH; ++k) cv += sh[k] * cond_w2[k * HH + h];

  tcvec[b * HH + h] = tv + cv;
}

// ---------------- diffusion noising ----------------
__global__ void noise_kernel(const float* __restrict__ node_features, const float* __restrict__ positions,
                             const float* __restrict__ fnoise, const float* __restrict__ pnoise,
                             const float* __restrict__ sab,
                             float* __restrict__ nf, float* __restrict__ npos) {
  const int idx = blockIdx.x * blockDim.x + threadIdx.x;  // over B*N*10
  if (idx >= BB * NN * 10) return;
  const int b = idx / (NN * 10);
  const int r = idx % (NN * 10);
  const int i = r / 10;
  const int f = r % 10;
  const float sa = sab[2 * b + 0], sb = sab[2 * b + 1];
  if (f < NDIM) {
    const int o = (b * NN + i) * NDIM + f;
    nf[o] = sa * node_features[o] + sb * fnoise[o];
  } else {
    const int o = (b * NN + i) * 2 + (f - NDIM);
    npos[o] = sa * positions[o] + sb * pnoise[o];
  }
}

// ---------------- node-embedding MLP layer 1 (K=10, naive) ----------------
__global__ void nodep_hidden_kernel(const float* __restrict__ nf, const float* __restrict__ npos,
                                    const float* __restrict__ w1, const float* __restrict__ b1,
                                    float* __restrict__ out) {
  const int idx = blockIdx.x * blockDim.x + threadIdx.x;  // over BN*H
  if (idx >= BN * HH) return;
  const int m = idx / HH, h = idx % HH;
  float acc = b1[h];
  for (int k = 0; k < NDIM; ++k) acc += nf[m * NDIM + k] * w1[k * HH + h];
  for (int k = 0; k < 2; ++k) acc += npos[m * 2 + k] * w1[(NDIM + k) * HH + h];
  out[idx] = silu_f(acc);
}

// ---------------- pairwise distances ----------------
__global__ void dist_kernel(const float* __restrict__ npos, float* __restrict__ dist) {
  const int idx = blockIdx.x * blockDim.x + threadIdx.x;  // over B*N*N
  if (idx >= BB * NN * NN) return;
  const int b = idx / (NN * NN);
  const int r = idx % (NN * NN);
  const int i = r / NN, j = r % NN;
  const float dx = npos[(b * NN + i) * 2 + 0] - npos[(b * NN + j) * 2 + 0];
  const float dy = npos[(b * NN + i) * 2 + 1] - npos[(b * NN + j) * 2 + 1];
  dist[idx] = sqrtf(dx * dx + dy * dy + 1e-12f);
}

// ---------------- dominant O(N^2 * H) edge silu accumulation ----------------
// Two i-rows per block: each aj load feeds two independent silu chains.
// hsum[b,i,h] = mask_i * sum_j mask_j * silu(ai[b,i,h] + aj[b,j,h] + d_ij*wd[h] + eb1[h])
// cnt[b,i]   = mask_i * sum_j mask_j
__global__ void edge_kernel(const float* __restrict__ ai, const float* __restrict__ aj,
                            const float* __restrict__ dist, const float* __restrict__ mask,
                            const float* __restrict__ wd, const float* __restrict__ eb1,
                            float* __restrict__ hsum, float* __restrict__ cnt) {
  const int pair = blockIdx.x;            // 0 .. B*N/2 - 1
  const int b = pair / (NN / 2);
  const int i0 = (pair % (NN / 2)) * 2;   // rows i0, i0+1
  const int h = threadIdx.x;              // 0..127
  __shared__ float sd[2 * NN];
  __shared__ float sm[NN];
  const size_t dbase = ((size_t)(b * NN + i0)) * NN;  // rows i0,i0+1 are contiguous
  for (int r = h; r < 2 * NN; r += HH) sd[r] = dist[dbase + r];
  for (int j = h; j < NN; j += HH) sm[j] = mask[b * NN + j];
  __syncthreads();

  const float eb = eb1[h];
  const float wdh = wd[h];
  const float ai0 = ai[(b * NN + i0) * HH + h] + eb;
  const float ai1 = ai[(b * NN + i0 + 1) * HH + h] + eb;
  const float* __restrict__ ajb = aj + (size_t)b * NN * HH;
  float acc0 = 0.f, acc1 = 0.f, cm = 0.f;
#pragma unroll 4
  for (int j = 0; j < NN; ++j) {
    const float av = ajb[j * HH + h];
    const float mj = sm[j];
    const float g0 = ai0 + av + sd[j] * wdh;
    const float g1 = ai1 + av + sd[NN + j] * wdh;
    acc0 += mj * (g0 / (1.0f + __expf(-g0)));
    acc1 += mj * (g1 / (1.0f + __expf(-g1)));
    cm += mj;
  }
  const float mi0 = mask[b * NN + i0];
  const float mi1 = mask[b * NN + i0 + 1];
  hsum[(b * NN + i0) * HH + h] = mi0 * acc0;
  hsum[(b * NN + i0 + 1) * HH + h] = mi1 * acc1;
  if (h == 0) {
    cnt[b * NN + i0] = mi0 * cm;
    cnt[b * NN + i0 + 1] = mi1 * cm;
  }
}

// ---------------- noise-loss reduction ----------------
__global__ void loss_reduce_kernel(const float* __restrict__ pred_fn, const float* __restrict__ fnoise,
                                   const float* __restrict__ pred_pn, const float* __restrict__ pnoise,
                                   const float* __restrict__ mask, float* __restrict__ accs) {
  const int r = blockIdx.x * blockDim.x + threadIdx.x;  // 0..BN-1
  float fl = 0.f, pl = 0.f, mm = 0.f;
  if (r < BN) {
    const float m = mask[r];
    mm = m;
    for (int k = 0; k < NDIM; ++k) {
      const float d = pred_fn[r * NDIM + k] - fnoise[r * NDIM + k];
      fl += d * d;
    }
    for (int k = 0; k < 2; ++k) {
      const float d = pred_pn[r * 2 + k] - pnoise[r * 2 + k];
      pl += d * d;
    }
    fl *= m;
    pl *= m;
  }
  __shared__ float s0[128], s1[128], s2[128];
  const int t = threadIdx.x;
  s0[t] = fl; s1[t] = pl; s2[t] = mm;
  __syncthreads();
  for (int off = 64; off > 0; off >>= 1) {
    if (t < off) { s0[t] += s0[t + off]; s1[t] += s1[t + off]; s2[t] += s2[t + off]; }
    __syncthreads();
  }
  if (t == 0) {
    atomicAdd(&accs[0], s0[0]);
    atomicAdd(&accs[1], s1[0]);
    atomicAdd(&accs[2], s2[0]);
  }
}

// ---------------- masked mean pooling ----------------
__global__ void gemb_kernel(const float* __restrict__ state, const float* __restrict__ mask,
                            float* __restrict__ gemb) {
  const int b = blockIdx.x;
  const int h = threadIdx.x;
  float acc = 0.f, msum = 0.f;
  for (int i = 0; i < NN; ++i) {
    const float m = mask[b * NN + i];
    acc += state[(b * NN + i) * HH + h] * m;
    msum += m;
  }
  gemb[b * HH + h] = acc / fmaxf(msum, 1.0f);
}

// ---------------- property head + final loss assembly ----------------
__global__ void final_kernel(const float* __restrict__ gemb, const float* __restrict__ targets,
                             const float* __restrict__ pw,
                             const float* __restrict__ w1, const float* __restrict__ b1,
                             const float* __restrict__ w2, const float* __restrict__ b2,
                             const float* __restrict__ w3, const float* __restrict__ b3,
                             const float* __restrict__ accs, float* __restrict__ out) {
  const int h = threadIdx.x;  // 0..127
  __shared__ float sg[HH], s1[HH], s2[HH / 2], sp[BB * 4];
  for (int b = 0; b < BB; ++b) {
    sg[h] = gemb[b * HH + h];
    __syncthreads();
    float a1 = b1[h];
    for (int k = 0; k < HH; ++k) a1 += sg[k] * w1[k * HH + h];
    __syncthreads();
    s1[h] = silu_f(a1);
    __syncthreads();
    if (h < HH / 2) {
      float a2 = b2[h];
      for (int k = 0; k < HH; ++k) a2 += s1[k] * w2[k * (HH / 2) + h];
      s2[h] = silu_f(a2);
    }
    __syncthreads();
    if (h < 4) {
      float a3 = b3[h];
      for (int k = 0; k < HH / 2; ++k) a3 += s2[k] * w3[k * 4 + h];
      const float d = a3 - targets[b * 4 + h];
      sp[b * 4 + h] = d * d * pw[h];
    }
    __syncthreads();
  }
  if (h == 0) {
    float pl = 0.f;
    for (int j = 0; j < BB * 4; ++j) pl += sp[j];
    pl /= (float)(BB * 4);
    const float msum = fmaxf(accs[2], 1.0f);
    const float nl = accs[0] / msum + accs[1] / msum;
    out[0] = nl;
    out[1] = pl;
    out[2] = nl + pl;
  }
}

// =======================================================================================
extern "C" void kernel_launch(void* const* d_in, const int* in_sizes, int n_in,
                              void* d_out, int out_size, void* d_ws, size_t ws_size,
                              hipStream_t stream) {
  const float* node_features = (const float*)d_in[0];
  const float* positions     = (const float*)d_in[1];
  const float* mask          = (const float*)d_in[2];
  const float* condition     = (const float*)d_in[3];
  const float* targets       = (const float*)d_in[4];
  const float* prop_weights  = (const float*)d_in[5];
  const float* feature_noise = (const float*)d_in[6];
  const float* position_noise= (const float*)d_in[7];
  const int*   timesteps     = (const int*)d_in[8];
  const float* time_w1 = (const float*)d_in[9],  *time_b1 = (const float*)d_in[10];
  const float* time_w2 = (const float*)d_in[11], *time_b2 = (const float*)d_in[12];
  const float* cond_w1 = (const float*)d_in[13], *cond_b1 = (const float*)d_in[14];
  const float* cond_w2 = (const float*)d_in[15], *cond_b2 = (const float*)d_in[16];
  const float* nodep_w1 = (const float*)d_in[17], *nodep_b1 = (const float*)d_in[18];
  const float* nodep_w2 = (const float*)d_in[19], *nodep_b2 = (const float*)d_in[20];
  const float* edge_w1 = (const float*)d_in[21], *edge_b1 = (const float*)d_in[22];
  const float* edge_w2 = (const float*)d_in[23], *edge_b2 = (const float*)d_in[24];
  const float* nodem_w1 = (const float*)d_in[25], *nodem_b1 = (const float*)d_in[26];
  const float* nodem_w2 = (const float*)d_in[27], *nodem_b2 = (const float*)d_in[28];
  const float* feat_w1 = (const float*)d_in[29], *feat_b1 = (const float*)d_in[30];
  const float* feat_w2 = (const float*)d_in[31], *feat_b2 = (const float*)d_in[32];
  const float* pos_w1 = (const float*)d_in[33], *pos_b1 = (const float*)d_in[34];
  const float* pos_w2 = (const float*)d_in[35], *pos_b2 = (const float*)d_in[36];
  const float* prop_w1 = (const float*)d_in[37], *prop_b1 = (const float*)d_in[38];
  const float* prop_w2 = (const float*)d_in[39], *prop_b2 = (const float*)d_in[40];
  const float* prop_w3 = (const float*)d_in[41], *prop_b3 = (const float*)d_in[42];
  float* out = (float*)d_out;

  // workspace layout (floats)
  float* w = (float*)d_ws;
  float* state = w;   w += BN * HH;
  float* tmp   = w;   w += BN * HH;
  float* ai    = w;   w += BN * HH;
  float* aj    = w;   w += BN * HH;
  float* hsum  = w;   w += BN * HH;
  float* agg   = w;   w += BN * HH;
  float* nf    = w;   w += BN * NDIM;
  float* npos  = w;   w += BN * 2;
  float* dist  = w;   w += (size_t)BB * NN * NN;
  float* cnt   = w;   w += BN;
  float* tcvec = w;   w += BB * HH;
  float* sab   = w;   w += BB * 2;
  float* pred_fn = w; w += BN * NDIM;
  float* pred_pn = w; w += BN * 2;
  float* gemb  = w;   w += BB * HH;
  float* accs  = w;   w += 4;

  hipMemsetAsync(accs, 0, 4 * sizeof(float), stream);

  // time/cond embeddings + diffusion scale factors
  prep_small_kernel<<<BB, HH, 0, stream>>>(timesteps, condition, time_w1, time_b1, time_w2,
                                           time_b2, cond_w1, cond_b1, cond_w2, cond_b2, sab,
                                           tcvec);
  noise_kernel<<<(BB * NN * 10 + 255) / 256, 256, 0, stream>>>(
      node_features, positions, feature_noise, position_noise, sab, nf, npos);

  const int gsq = BN / 16;  // 4 waves/block cover all 128 output columns

  // state init: mlp2(cat(nf,npos)) + tvec + cvec (tcvec folded into GEMM epilogue)
  nodep_hidden_kernel<<<(BN * HH + 255) / 256, 256, 0, stream>>>(nf, npos, nodep_w1, nodep_b1, tmp);
  gemm16x32_wmma<EPI_BIAS_ADDB><<<gsq, 128, 0, stream>>>(tmp, nodep_w2, state, nodep_b2,
                                                         nullptr, tcvec, BN, HH, HH);

  dist_kernel<<<(BB * NN * NN + 255) / 256, 256, 0, stream>>>(npos, dist);

  for (int l = 0; l < LL; ++l) {
    const float* ew1a = edge_w1 + (size_t)l * (2 * HH + 1) * HH;
    const float* ew1b = ew1a + HH * HH;
    const float* wd   = ew1a + 2 * HH * HH;
    const float* eb1  = edge_b1 + l * HH;
    const float* ew2  = edge_w2 + (size_t)l * HH * HH;
    const float* eb2  = edge_b2 + l * HH;
    const float* nw1a = nodem_w1 + (size_t)l * 2 * HH * HH;
    const float* nw1b = nw1a + HH * HH;
    const float* nb1  = nodem_b1 + l * HH;
    const float* nw2  = nodem_w2 + (size_t)l * HH * HH;
    const float* nb2  = nodem_b2 + l * HH;

    gemm16x32_wmma<EPI_NONE><<<gsq, 128, 0, stream>>>(state, ew1a, ai, nullptr, nullptr, nullptr, BN, HH, HH);
    gemm16x32_wmma<EPI_NONE><<<gsq, 128, 0, stream>>>(state, ew1b, aj, nullptr, nullptr, nullptr, BN, HH, HH);
    edge_kernel<<<BN / 2, HH, 0, stream>>>(ai, aj, dist, mask, wd, eb1, hsum, cnt);
    gemm16x32_wmma<EPI_EDGE_AGG><<<gsq, 128, 0, stream>>>(hsum, ew2, agg, eb2, nullptr, cnt, BN, HH, HH);
    gemm16x32_wmma<EPI_BIAS><<<gsq, 128, 0, stream>>>(state, nw1a, tmp, nb1, nullptr, nullptr, BN, HH, HH);
    gemm16x32_wmma<EPI_ACC_SILU><<<gsq, 128, 0, stream>>>(agg, nw1b, tmp, nullptr, tmp, nullptr, BN, HH, HH);
    gemm16x32_wmma<EPI_ADD_SCALED><<<gsq, 128, 0, stream>>>(tmp, nw2, state, nb2, state, mask, BN, HH, HH);
  }

  // output heads
  gemm16x32_wmma<EPI_BIAS_SILU><<<gsq, 128, 0, stream>>>(state, feat_w1, tmp, feat_b1, nullptr,
                                                         nullptr, BN, HH, HH);
  gemm16_narrow_wmma<EPI_BIAS><<<BN / 16, 32, 0, stream>>>(tmp, feat_w2, pred_fn, feat_b2,
                                                           BN, HH, NDIM);
  gemm16x32_wmma<EPI_BIAS_SILU><<<gsq, 128, 0, stream>>>(state, pos_w1, agg, pos_b1, nullptr,
                                                         nullptr, BN, HH, HH);
  gemm16_narrow_wmma<EPI_BIAS><<<BN / 16, 32, 0, stream>>>(agg, pos_w2, pred_pn, pos_b2,
                                                           BN, HH, 2);

  loss_reduce_kernel<<<(BN + 127) / 128, 128, 0, stream>>>(pred_fn, feature_noise, pred_pn,
                                                           position_noise, mask, accs);
  gemb_kernel<<<BB, HH, 0, stream>>>(state, mask, gemb);
  final_kernel<<<1, HH, 0, stream>>>(gemb, targets, prop_weights, prop_w1, prop_b1, prop_w2,
                                     prop_b2, prop_w3, prop_b3, accs, out);
}